// ReweightLoss_43430709297431
// MI455X (gfx1250) — compile-verified
//
#include <hip/hip_runtime.h>
#include <stdint.h>

#define TPB 128
#define MAXC_LDS 1024

// ---------- wave32 / block reductions ----------
__device__ __forceinline__ float wave_redmax(float v) {
#pragma unroll
  for (int o = 16; o > 0; o >>= 1) v = fmaxf(v, __shfl_xor(v, o, 32));
  return v;
}
__device__ __forceinline__ float wave_redsum(float v) {
#pragma unroll
  for (int o = 16; o > 0; o >>= 1) v += __shfl_xor(v, o, 32);
  return v;
}
__device__ __forceinline__ float block_redmax(float v, float* sc) {
  const int t = threadIdx.x, lane = t & 31, wv = t >> 5, nw = TPB >> 5;
  v = wave_redmax(v);
  __syncthreads();                 // protect scratch reuse across calls
  if (lane == 0) sc[wv] = v;
  __syncthreads();
  if (wv == 0) {
    float u = (lane < nw) ? sc[lane] : -__builtin_inff();
    u = wave_redmax(u);
    if (lane == 0) sc[0] = u;
  }
  __syncthreads();
  return sc[0];
}
__device__ __forceinline__ float block_redsum(float v, float* sc) {
  const int t = threadIdx.x, lane = t & 31, wv = t >> 5, nw = TPB >> 5;
  v = wave_redsum(v);
  __syncthreads();
  if (lane == 0) sc[wv] = v;
  __syncthreads();
  if (wv == 0) {
    float u = (lane < nw) ? sc[lane] : 0.0f;
    u = wave_redsum(u);
    if (lane == 0) sc[0] = u;
  }
  __syncthreads();
  return sc[0];
}

// ---------- kernel 1: tiled transpose T (CxC) -> Tt ----------
__global__ __launch_bounds__(256) void transpose_kernel(const float* __restrict__ A,
                                                        float* __restrict__ At, int C) {
  __shared__ float tile[32][33];
  const int bx = blockIdx.x << 5;  // col base in A
  const int by = blockIdx.y << 5;  // row base in A
  for (int i = threadIdx.y; i < 32; i += 8) {
    int rr = by + i, cc = bx + (int)threadIdx.x;
    if (rr < C && cc < C) tile[i][threadIdx.x] = A[(size_t)rr * C + cc];
  }
  __syncthreads();
  for (int i = threadIdx.y; i < 32; i += 8) {
    int rr = bx + i, cc = by + (int)threadIdx.x;   // At[col][row] = A[row][col]
    if (rr < C && cc < C) At[(size_t)rr * C + cc] = tile[threadIdx.x][i];
  }
}

// ---------- kernel 2: one block per row ----------
__global__ __launch_bounds__(TPB) void reweight_row_kernel(
    const float* __restrict__ logits, const int* __restrict__ target,
    const float* __restrict__ T, const float* __restrict__ Tt,
    float* __restrict__ partials, int C, int useTt) {
  __shared__ float row[MAXC_LDS];
  __shared__ float sc[8];
  const int r = blockIdx.x;
  const int t = threadIdx.x;
  const float* __restrict__ g = logits + (size_t)r * (size_t)C;
  const float* src;

  const bool stageLds  = (C <= MAXC_LDS);
  const bool asyncPath = stageLds && ((C & 3) == 0);

  // (1) issue async global->LDS copies of this row (ASYNCcnt path, 16B/lane/issue)
  if (asyncPath) {
    const unsigned ldsBase = (unsigned)(uintptr_t)(&row[0]);  // low 32b = LDS offset
    const int nch = C >> 2;
    for (int ch = t; ch < nch; ch += TPB) {
      unsigned lds = ldsBase + ((unsigned)ch << 4);
      const float* ga = g + (ch << 2);
      asm volatile("global_load_async_to_lds_b128 %0, %1, off"
                   :: "v"(lds), "v"(ga) : "memory");
    }
  }

  // (2) overlap the DMA with independent work: target load + Tt-row prefetch
  const int y = target[r];
  const float* __restrict__ tcol = useTt ? (Tt + (size_t)y * (size_t)C) : (T + y);
  if (useTt) {
    const int lines = (C + 31) >> 5;                // one prefetch per 128B line
    if (t < lines) __builtin_prefetch(tcol + (t << 5), 0, 0);
  }

  // (3) now wait for our wave's async copies, then barrier for the whole block
  if (stageLds) {
    if (asyncPath) {
#if __has_builtin(__builtin_amdgcn_s_wait_asynccnt)
      __builtin_amdgcn_s_wait_asynccnt(0);
#else
      asm volatile("s_wait_asynccnt 0" ::: "memory");
#endif
    } else {
      for (int c = t; c < C; c += TPB) row[c] = g[c];
    }
    __syncthreads();
    src = row;
  } else {
    src = g;  // fallback: operate straight from global (L2-resident)
  }

  // pass 1: row max
  float m = -__builtin_inff();
  for (int c = t; c < C; c += TPB) m = fmaxf(m, src[c]);
  m = block_redmax(m, sc);

  // pass 2: Z = sum exp, d = sum exp * T[c,y]
  float z = 0.0f, d = 0.0f;
  if (useTt) {
    for (int c = t; c < C; c += TPB) {
      float e = expf(src[c] - m);
      z += e;
      d = fmaf(e, tcol[c], d);
    }
  } else {
    for (int c = t; c < C; c += TPB) {
      float e = expf(src[c] - m);
      z += e;
      d = fmaf(e, tcol[(size_t)c * (size_t)C], d);
    }
  }
  z = block_redsum(z, sc);
  d = block_redsum(d, sc);

  if (t == 0) {
    float xy = src[y];
    float beta = expf(xy - m) / d;          // Z cancels in pro1/pro2
    float ce   = logf(z) - (xy - m);        // -log_softmax at target
    partials[r] = beta * ce;
  }
}

// ---------- kernel 3: deterministic final reduction ----------
__global__ __launch_bounds__(256) void final_reduce_kernel(const float* __restrict__ p,
                                                           float* __restrict__ outp, int n) {
  __shared__ float s[256];
  float acc = 0.0f;
  for (int i = threadIdx.x; i < n; i += 256) acc += p[i];
  s[threadIdx.x] = acc;
  __syncthreads();
  for (int o = 128; o > 0; o >>= 1) {
    if ((int)threadIdx.x < o) s[threadIdx.x] += s[threadIdx.x + o];
    __syncthreads();
  }
  if (threadIdx.x == 0) outp[0] = s[0];
}

extern "C" void kernel_launch(void* const* d_in, const int* in_sizes, int n_in,
                              void* d_out, int out_size, void* d_ws, size_t ws_size,
                              hipStream_t stream) {
  const float* logits = (const float*)d_in[0];   // [B, C] f32
  const int*   target = (const int*)d_in[1];     // [B]
  const float* T      = (const float*)d_in[2];   // [C, C] f32
  const int B = in_sizes[1];
  const int C = (B > 0) ? (in_sizes[0] / B) : 0;

  // workspace layout: [0, B*4) partials ; [align256, +C*C*4) transposed T
  float* partials = (float*)d_ws;
  size_t ttOff = (((size_t)B * sizeof(float)) + 255) & ~(size_t)255;
  size_t need  = ttOff + (size_t)C * (size_t)C * sizeof(float);
  int useTt = (ws_size >= need) ? 1 : 0;
  float* Tt = (float*)((char*)d_ws + ttOff);

  if (useTt) {
    dim3 gb((C + 31) / 32, (C + 31) / 32);
    transpose_kernel<<<gb, dim3(32, 8), 0, stream>>>(T, Tt, C);
  }
  reweight_row_kernel<<<B, TPB, 0, stream>>>(logits, target, T, Tt, partials, C, useTt);
  final_reduce_kernel<<<1, 256, 0, stream>>>(partials, (float*)d_out, B);
}